// AtomicOrbitals_8014408974508
// MI455X (gfx1250) — compile-verified
//
#include <hip/hip_runtime.h>
#include <hip/hip_bf16.h>

typedef __attribute__((ext_vector_type(2))) float v2f;
typedef __attribute__((ext_vector_type(8))) float v8f;

#define NBATCH 1024
#define NELEC  64
#define NBAS   256
#define NORB   128
#define NROWS  (NBATCH * NELEC)
#define TILE_M 16
#define KSTRIDE 260   // 256 + 4 pad: avoids LDS bank conflicts on A-fragment reads

__global__ __launch_bounds__(256)
void AtomicOrbitals_wmma_kernel(
    const float* __restrict__ input,          // [NROWS, 3]
    const float* __restrict__ atom_coords,    // [16, 3]
    const float* __restrict__ bas_exp,        // [NBAS]
    const float* __restrict__ bas_coeffs,     // [NBAS]
    const float* __restrict__ norm_cst,       // [NBAS]
    const int*   __restrict__ bas_l,          // [NBAS]
    const int*   __restrict__ bas_m,          // [NBAS]
    const int*   __restrict__ bas_atom_index, // [NBAS]
    const int*   __restrict__ index_ctr,      // [NBAS]
    float*       __restrict__ out)            // [NROWS, NORB]
{
    __shared__ float sbx[NBAS], sby[NBAS], sbz[NBAS], sbe[NBAS], scc[NBAS];
    __shared__ int   sl[NBAS], smm[NBAS], sic[NBAS];
    __shared__ float tile[TILE_M * KSTRIDE];  // bas[m][k] for this M-tile

    const int t = threadIdx.x;

    // ---- Preload per-basis constants into LDS (256 threads, 1 each) ----
    {
        int ai = bas_atom_index[t];
        sbx[t] = atom_coords[ai * 3 + 0];
        sby[t] = atom_coords[ai * 3 + 1];
        sbz[t] = atom_coords[ai * 3 + 2];
        sbe[t] = bas_exp[t];
        scc[t] = norm_cst[t] * bas_coeffs[t];
        sl[t]  = bas_l[t];
        smm[t] = bas_m[t];
        sic[t] = index_ctr[t];
    }
    __syncthreads();

    // ---- Phase A: bas[m, k] for 16 rows x 256 basis fns ----
    // r^l * Y cancels the 1/r (l=1) and 1/r^2 (l=2) factors exactly:
    //   bas = cc * exp(-a*r2) * Ypoly     (no sqrt, no divide)
    const int   m   = t & 15;
    const int   row = blockIdx.x * TILE_M + m;
    const float ex = input[row * 3 + 0];
    const float ey = input[row * 3 + 1];
    const float ez = input[row * 3 + 2];

    const float C0   = 0.2820948f;
    const float C1   = 0.4886025f;
    const float C2XY = 1.0925484f;
    const float C2Z2 = 0.31539156f;
    const float C2D  = 0.5462742f;

    #pragma unroll
    for (int i = 0; i < 16; ++i) {
        int   k  = (t >> 4) + i * 16;
        float dx = ex - sbx[k];
        float dy = ey - sby[k];
        float dz = ez - sbz[k];
        float r2 = dx * dx + dy * dy + dz * dz;
        float e  = __expf(-sbe[k] * r2);
        int   l  = sl[k];
        int   mq = smm[k];
        float Y;
        if (l == 0) {
            Y = C0;
        } else if (l == 1) {
            Y = C1 * (mq < 0 ? dy : (mq == 0 ? dz : dx));
        } else {
            Y = (mq == -2) ? C2XY * dx * dy
              : (mq == -1) ? C2XY * dy * dz
              : (mq ==  0) ? C2Z2 * (2.0f * dz * dz - dx * dx - dy * dy)
              : (mq ==  1) ? C2XY * dz * dx
              :              C2D  * (dx * dx - dy * dy);
        }
        tile[m * KSTRIDE + k] = scc[k] * e * Y;
    }
    __syncthreads();

    // ---- Phase B: ao[16 x 128] = bas[16 x 256] * P[256 x 128] via fp32 WMMA ----
    // P[k][n] = (index_ctr[k] == n) ? 1 : 0, generated on the fly (exact in f32).
    // Wave w owns output columns [16w, 16w+16).
    const int lane = t & 31;
    const int wave = t >> 5;
    const int half = lane >> 4;   // K-half selector for A/B fragments
    const int ln   = lane & 15;   // M (for A) / N (for B,C) within tile
    const int n    = wave * 16 + ln;

    v8f c = {};
    #pragma unroll
    for (int k0 = 0; k0 < NBAS; k0 += 4) {
        int ka = k0 + half * 2;
        v2f a, b;
        // A 16x4 f32: lanes 0-15 -> M=lane, K=k0+{0,1}; lanes 16-31 -> K=k0+{2,3}
        a.x = tile[ln * KSTRIDE + ka];
        a.y = tile[ln * KSTRIDE + ka + 1];
        // B 4x16 f32: lanes 0-15 -> N=lane, K=k0+{0,1}; lanes 16-31 -> K=k0+{2,3}
        b.x = (sic[ka]     == n) ? 1.0f : 0.0f;
        b.y = (sic[ka + 1] == n) ? 1.0f : 0.0f;
        c = __builtin_amdgcn_wmma_f32_16x16x4_f32(
                /*neg_a=*/false, a, /*neg_b=*/false, b,
                /*c_mod=*/(short)0, c, /*reuse_a=*/false, /*reuse_b=*/false);
    }

    // C/D 16x16 f32: VGPR j -> M = j + half*8, N = ln
    const int rbase = blockIdx.x * TILE_M + half * 8;
    #pragma unroll
    for (int j = 0; j < 8; ++j) {
        out[(size_t)(rbase + j) * NORB + n] = c[j];
    }
}

extern "C" void kernel_launch(void* const* d_in, const int* in_sizes, int n_in,
                              void* d_out, int out_size, void* d_ws, size_t ws_size,
                              hipStream_t stream) {
    const float* input          = (const float*)d_in[0];
    const float* atom_coords    = (const float*)d_in[1];
    const float* bas_exp        = (const float*)d_in[2];
    const float* bas_coeffs     = (const float*)d_in[3];
    const float* norm_cst       = (const float*)d_in[4];
    // d_in[5] = bas_n (f32) — redundant with bas_l, unused
    const int*   bas_l          = (const int*)d_in[6];
    const int*   bas_m          = (const int*)d_in[7];
    const int*   bas_atom_index = (const int*)d_in[8];
    const int*   index_ctr      = (const int*)d_in[9];
    float*       out            = (float*)d_out;

    dim3 grid(NROWS / TILE_M);  // 4096 workgroups, one 16-row tile each
    dim3 block(256);            // 8 wave32s
    AtomicOrbitals_wmma_kernel<<<grid, block, 0, stream>>>(
        input, atom_coords, bas_exp, bas_coeffs, norm_cst,
        bas_l, bas_m, bas_atom_index, index_ctr, out);
}